// LocallyConnected1d_88347477279631
// MI455X (gfx1250) — compile-verified
//
#include <hip/hip_runtime.h>

#define BATCH 32
#define CIN   64
#define COUT  64
#define IN_W  2048
#define KW    3
#define CK    (CIN * KW)          // 192 reduction length (k-major index: j = k*64 + c)
#define WBLK  8                   // output positions per block
#define XS_W  (WBLK + 2)          // slab width incl. halo = 10

#define XS_BSTRIDE 646            // per-batch stride (dwords): even (b64-aligned), bank spread
#define XS_SIZE (BATCH * XS_BSTRIDE)        // 20672 floats
#define WS_ROW  196               // weight row stride: 16B-aligned stores, conflict-free b64 reads
#define WS_SIZE (COUT * WS_ROW)             // 12544 floats per buffer
#define CS_ROW  9                 // padded C staging stride
#define CS_SIZE (BATCH * COUT * CS_ROW)     // 18432 floats

#define LDS_FLOATS (XS_SIZE + 2 * WS_SIZE + CS_SIZE)
#define LDS_BYTES  (LDS_FLOATS * 4)         // 256,768 B <= 320 KB

#define NTHREADS 512              // 16 wave32 waves: 8 compute + 8 staging

typedef __attribute__((ext_vector_type(2))) float v2f;
typedef __attribute__((ext_vector_type(4))) float v4f;
typedef __attribute__((ext_vector_type(8))) float v8f;

// ---- weight staging: global (o,c,k) -> LDS k-major rows ws[o*196 + k*64 + c]
// 256 staging threads, 12288 floats: thread handles 12 groups of 4 consecutive j.
// Non-temporal loads: the weight stream is read-once, keep it out of L2's way.
__device__ __forceinline__ void wstage(const float* __restrict__ wsrc,
                                       float* __restrict__ dst, int t2) {
    v4f r[12];
    #pragma unroll
    for (int jj = 0; jj < 12; ++jj) {
        const int f = jj * 256 + t2;
        const int o = f / 48;
        const int q = f % 48;                    // j0 = 4q : k = q>>4, c0 = 4*(q&15)
        const float* g = wsrc + o * CK + (q >> 4) + 12 * (q & 15);
        r[jj].x = __builtin_nontemporal_load(g + 0);
        r[jj].y = __builtin_nontemporal_load(g + 3);
        r[jj].z = __builtin_nontemporal_load(g + 6);
        r[jj].w = __builtin_nontemporal_load(g + 9);
    }
    #pragma unroll
    for (int jj = 0; jj < 12; ++jj) {
        const int f = jj * 256 + t2;
        const int o = f / 48;
        const int q = f % 48;
        *(v4f*)(dst + o * WS_ROW + 4 * q) = r[jj];
    }
}

// ---- one 32x64x192 position GEMM for this wave's 16x16 tile
__device__ __forceinline__ v8f tile_gemm(const float* __restrict__ baseA,
                                         const float* __restrict__ baseB) {
    v8f acc = {};
    #pragma unroll
    for (int s = 0; s < 48; ++s) {
        const int offA = ((s >> 4) << 6) + ((4 * s) & 63);   // compile-time immediate
        v2f a   = *(const v2f*)(baseA + offA);
        v2f bfr = *(const v2f*)(baseB + 4 * s);
        acc = __builtin_amdgcn_wmma_f32_16x16x4_f32(
                  false, a, false, bfr, (short)0, acc, false, false);
    }
    return acc;
}

__global__ __launch_bounds__(NTHREADS)
void lc1d_wmma_f32(
    const float* __restrict__ x,     // [32][64][2048]
    const float* __restrict__ wgt,   // [2048][64][64][3]  (o, c, k) per position
    const float* __restrict__ bias,  // [64][2048]
    float* __restrict__ out)         // [32][64][2048]
{
    extern __shared__ float smem[];
    float* xs  = smem;                          // x slab
    float* wsA = smem + XS_SIZE;                // weight double buffer (k-major rows)
    float* cs  = smem + XS_SIZE + 2 * WS_SIZE;  // C staging

    const int tid  = threadIdx.x;
    const int lane = tid & 31;
    const int wid  = tid >> 5;
    const int hl   = lane >> 4;      // half-wave: holds K+2/K+3 per ISA A/B layouts
    const int lidx = lane & 15;
    const int w0   = blockIdx.x * WBLK;
    const bool is_compute = (wid < 8);           // wave-uniform role split

    if (is_compute) {
        // ---- compute waves fill x slab: xs[b*646 + wabs*64 + c] = x[b][c][w0-1+wabs]
        for (int p = tid; p < BATCH * CIN; p += 256) {
            const int b = p >> 6;
            const int c = p & 63;
            const float* xp = x + ((size_t)b * CIN + c) * IN_W + (w0 - 1);
            float* xd = xs + b * XS_BSTRIDE + c;
            #pragma unroll
            for (int wa = 0; wa < XS_W; ++wa) {
                const int wg = w0 - 1 + wa;
                float v = 0.0f;
                if (wg >= 0 && wg < IN_W) v = xp[wa];
                xd[wa * 64] = v;
            }
        }
    } else {
        // ---- staging waves prime buffer 0 with position 0's weights
        wstage(wgt + (size_t)w0 * (COUT * CK), wsA, tid - 256);
    }
    __syncthreads();

    const int mt  = wid & 1;         // batch 16-tile      (compute waves only)
    const int nt  = (wid >> 1) & 3;  // out-channel 16-tile
    const int b_m = mt * 16 + lidx;
    const int n_g = nt * 16 + lidx;
    const float* baseA0 = xs + b_m * XS_BSTRIDE + 2 * hl;
    const int    baseBo = n_g * WS_ROW + 2 * hl;

    for (int wl = 0; wl < WBLK; ++wl) {
        if (is_compute) {
            v8f acc = tile_gemm(baseA0 + wl * 64, wsA + (wl & 1) * WS_SIZE + baseBo);
            #pragma unroll
            for (int rr = 0; rr < 8; ++rr) {
                const int m = mt * 16 + hl * 8 + rr;
                cs[(m * COUT + n_g) * CS_ROW + wl] = acc[rr];
            }
        } else if (wl + 1 < WBLK) {
            // staging waves stream next position's weights into the other buffer
            wstage(wgt + (size_t)(w0 + wl + 1) * (COUT * CK),
                   wsA + (((wl & 1) ^ 1) * WS_SIZE), tid - 256);
        }
        __syncthreads();
    }

    // ---- all 512 threads: bias add + non-temporal b128 stores of out[b][o][w0..w0+7]
    for (int p = tid; p < BATCH * COUT; p += NTHREADS) {
        const int b = p >> 6;
        const int o = p & 63;
        const float* bp = bias + (size_t)o * IN_W + w0;
        float*       op = out  + ((size_t)b * COUT + o) * IN_W + w0;
        const float* cp = cs + p * CS_ROW;
        v4f r0, r1;
        r0.x = cp[0] + bp[0]; r0.y = cp[1] + bp[1];
        r0.z = cp[2] + bp[2]; r0.w = cp[3] + bp[3];
        r1.x = cp[4] + bp[4]; r1.y = cp[5] + bp[5];
        r1.z = cp[6] + bp[6]; r1.w = cp[7] + bp[7];
        __builtin_nontemporal_store(r0, (v4f*)(op));
        __builtin_nontemporal_store(r1, (v4f*)(op + 4));
    }
}

extern "C" void kernel_launch(void* const* d_in, const int* in_sizes, int n_in,
                              void* d_out, int out_size, void* d_ws, size_t ws_size,
                              hipStream_t stream) {
    const float* x    = (const float*)d_in[0];
    const float* wgt  = (const float*)d_in[1];
    const float* bias = (const float*)d_in[2];
    float* out = (float*)d_out;

    static_assert(LDS_BYTES <= 320 * 1024, "LDS budget");
    hipFuncSetAttribute(reinterpret_cast<const void*>(lc1d_wmma_f32),
                        hipFuncAttributeMaxDynamicSharedMemorySize, LDS_BYTES);

    dim3 grid(IN_W / WBLK);   // 256 workgroups, one per 8 output positions
    dim3 block(NTHREADS);     // 16 wave32 waves: 8 compute + 8 staging
    lc1d_wmma_f32<<<grid, block, LDS_BYTES, stream>>>(x, wgt, bias, out);
}